// MoCo_21363167330882
// MI455X (gfx1250) — compile-verified
//
#include <hip/hip_runtime.h>
#include <hip/hip_bf16.h>

// ---------------------------------------------------------------------------
// MoCo forward for MI455X (gfx1250), fp32 WMMA (V_WMMA_F32_16X16X4_F32) path.
// B=1024, EMBED=2048, DIM=128, K=65536, NUM_INFO=1024, NUM_COARSE=100, T=0.07
// Compile-time shapes, register double-buffered (software pipelined) WMMA
// loops, 32x64 wave tiles on the big GEMMs.
// ---------------------------------------------------------------------------

typedef float v2f __attribute__((ext_vector_type(2)));
typedef float v8f __attribute__((ext_vector_type(8)));

#define NB      1024
#define NEMBED  2048
#define NDIM    128
#define NQ      65536
#define NINFO   1024
#define NCOARSE 100
#define INV_T   (1.0f / 0.07f)

__device__ __forceinline__ v8f wmma_f32(v2f a, v2f b, v8f c) {
    // D = A(16x4) * B(4x16) + C(16x16), fp32, wave32
    return __builtin_amdgcn_wmma_f32_16x16x4_f32(
        false, a, false, b, (short)0, c, false, false);
}

// ---------------------------------------------------------------------------
// C[M,N] = act(A[M,K] @ W[K,N] + bias)   (all row-major, shapes compile-time)
// block = 256 = 8 waves; wave tile = (16*MT) rows x 64 cols.
// grid = (ceil(N/512), M/(16*MT)).
// WGUARD: wave-level early-out for N < 512. EGUARD: per-element col guard
// (only the N=100 classifier; simple non-pipelined loop there).
// ---------------------------------------------------------------------------
template<int K, int N, bool RELU, int MT, bool WGUARD, bool EGUARD>
__global__ __launch_bounds__(256) void gemm_nn_t(const float* __restrict__ A,
                                                 const float* __restrict__ W,
                                                 const float* __restrict__ bias,
                                                 float* __restrict__ C)
{
    const int lane = threadIdx.x & 31;
    const int wave = threadIdx.x >> 5;
    const int row0 = blockIdx.y * (16 * MT);
    const int col0 = (blockIdx.x * 8 + wave) * 64;
    if (WGUARD && col0 >= N) return;               // wave-uniform: EXEC stays full

    const int m  = lane & 15;                      // A row / B col within tile
    const int kh = (lane >> 4) * 2;                // K sub-offset for this half
    const int rh = (lane >> 4) * 8;                // C row half
    v8f acc[MT][4] = {};

    const float* arow[MT];
#pragma unroll
    for (int r = 0; r < MT; ++r)
        arow[r] = A + (size_t)(row0 + r * 16 + m) * K + kh;

    if (EGUARD) {
        // cold path (classifier): simple guarded loop
        for (int k = 0; k < K; k += 4) {
            const v2f a = *(const v2f*)(arow[0] + k);
#pragma unroll
            for (int t = 0; t < 4; ++t) {
                const int col = col0 + t * 16 + m;
                v2f b;
                if (col < N) {
                    b.x = W[(size_t)(k + kh)     * N + col];
                    b.y = W[(size_t)(k + kh + 1) * N + col];
                } else { b.x = 0.f; b.y = 0.f; }
                acc[0][t] = wmma_f32(a, b, acc[0][t]);
            }
        }
    } else {
        // hot path: register double-buffered pipeline, last chunk peeled
        int col[4];
#pragma unroll
        for (int t = 0; t < 4; ++t) col[t] = col0 + t * 16 + m;

        v2f a_c[MT], b_c[4];
#pragma unroll
        for (int r = 0; r < MT; ++r) a_c[r] = *(const v2f*)(arow[r]);
        {
            const float* wp = W + (size_t)kh * N;
#pragma unroll
            for (int t = 0; t < 4; ++t) { b_c[t].x = wp[col[t]]; b_c[t].y = wp[N + col[t]]; }
        }
#pragma unroll 4
        for (int k = 0; k < K - 4; k += 4) {
            v2f a_n[MT], b_n[4];
#pragma unroll
            for (int r = 0; r < MT; ++r) a_n[r] = *(const v2f*)(arow[r] + k + 4);
            const float* wp = W + (size_t)(k + 4 + kh) * N;
#pragma unroll
            for (int t = 0; t < 4; ++t) { b_n[t].x = wp[col[t]]; b_n[t].y = wp[N + col[t]]; }
#pragma unroll
            for (int r = 0; r < MT; ++r)
#pragma unroll
                for (int t = 0; t < 4; ++t)
                    acc[r][t] = wmma_f32(a_c[r], b_c[t], acc[r][t]);
#pragma unroll
            for (int r = 0; r < MT; ++r) a_c[r] = a_n[r];
#pragma unroll
            for (int t = 0; t < 4; ++t) b_c[t] = b_n[t];
        }
#pragma unroll
        for (int r = 0; r < MT; ++r)
#pragma unroll
            for (int t = 0; t < 4; ++t)
                acc[r][t] = wmma_f32(a_c[r], b_c[t], acc[r][t]);
    }

    // epilogue
#pragma unroll
    for (int r = 0; r < MT; ++r)
#pragma unroll
        for (int t = 0; t < 4; ++t) {
            const int col = col0 + t * 16 + m;
            if (EGUARD && col >= N) continue;
            const float bv = bias[col];
#pragma unroll
            for (int v = 0; v < 8; ++v) {
                float x = acc[r][t][v] + bv;
                if (RELU) x = fmaxf(x, 0.f);
                C[(size_t)(row0 + r * 16 + rh + v) * N + col] = x;
            }
        }
}

// ---------------------------------------------------------------------------
// C[r, ccol0+c] = scale * sum_d A[r,d] * Bt[c,d]     (A:[M,KC], Bt:[N,KC])
// Used for l_neg (into d_out, ldc=65537, ccol0=1, scale=1/T) and scores.
// wave tile = (16*MT) rows x 64 cols; KC fully unrolled, pipelined.
// grid = (N/512, M/(16*MT)), block = 256.
// ---------------------------------------------------------------------------
template<int KC, int MT>
__global__ __launch_bounds__(256) void gemm_nt_t(const float* __restrict__ A,
                                                 const float* __restrict__ Bt,
                                                 float* __restrict__ C,
                                                 float scale, long long ldc, int ccol0)
{
    const int lane = threadIdx.x & 31;
    const int wave = threadIdx.x >> 5;
    const int row0 = blockIdx.y * (16 * MT);
    const int col0 = (blockIdx.x * 8 + wave) * 64;

    const int m  = lane & 15;
    const int kh = (lane >> 4) * 2;
    const int rh = (lane >> 4) * 8;
    v8f acc[MT][4] = {};

    const float* arow[MT];
#pragma unroll
    for (int r = 0; r < MT; ++r)
        arow[r] = A + (size_t)(row0 + r * 16 + m) * KC + kh;
    const float* brow[4];
#pragma unroll
    for (int t = 0; t < 4; ++t)
        brow[t] = Bt + (size_t)(col0 + t * 16 + m) * KC + kh;

    v2f a_c[MT], b_c[4];
#pragma unroll
    for (int r = 0; r < MT; ++r) a_c[r] = *(const v2f*)(arow[r]);
#pragma unroll
    for (int t = 0; t < 4; ++t) b_c[t] = *(const v2f*)(brow[t]);

#pragma unroll
    for (int k = 0; k < KC - 4; k += 4) {
        v2f a_n[MT], b_n[4];
#pragma unroll
        for (int r = 0; r < MT; ++r) a_n[r] = *(const v2f*)(arow[r] + k + 4);
#pragma unroll
        for (int t = 0; t < 4; ++t) b_n[t] = *(const v2f*)(brow[t] + k + 4);
#pragma unroll
        for (int r = 0; r < MT; ++r)
#pragma unroll
            for (int t = 0; t < 4; ++t)
                acc[r][t] = wmma_f32(a_c[r], b_c[t], acc[r][t]);
#pragma unroll
        for (int r = 0; r < MT; ++r) a_c[r] = a_n[r];
#pragma unroll
        for (int t = 0; t < 4; ++t) b_c[t] = b_n[t];
    }
#pragma unroll
    for (int r = 0; r < MT; ++r)
#pragma unroll
        for (int t = 0; t < 4; ++t)
            acc[r][t] = wmma_f32(a_c[r], b_c[t], acc[r][t]);

#pragma unroll
    for (int r = 0; r < MT; ++r)
#pragma unroll
        for (int t = 0; t < 4; ++t) {
            const long long col = col0 + t * 16 + m;
#pragma unroll
            for (int v = 0; v < 8; ++v)
                C[(long long)(row0 + r * 16 + rh + v) * ldc + ccol0 + col] =
                    acc[r][t][v] * scale;
        }
}

// ---------------------------------------------------------------------------
// dino: fused sim GEMM + mae/mask row reductions.
// Each wave covers 16 rows x 256 cols (4 passes of a pipelined 16x64 tile);
// per-row sums of {mask, mae*mask, mae*(1-mask)} reduced by xor-butterfly
// within each 16-lane half, accumulated in registers, 3 atomics/row/wave.
// grid = (NQ/2048, NB/16), block = 256.
// ---------------------------------------------------------------------------
__global__ __launch_bounds__(256) void dino_accum(const float* __restrict__ q,
                                                  const float* __restrict__ Qc,
                                                  const int*   __restrict__ label,
                                                  const int*   __restrict__ pseudo,
                                                  float* __restrict__ rowacc)
{
    const int lane = threadIdx.x & 31;
    const int wave = threadIdx.x >> 5;
    const int row0 = blockIdx.y * 16;
    const int colbase = (blockIdx.x * 8 + wave) * 256;

    const int m  = lane & 15;
    const int kh = (lane >> 4) * 2;
    const int rh = (lane >> 4) * 8;

    float aMsk[8], aMM[8], aMI[8];
    int   ps[8];
#pragma unroll
    for (int v = 0; v < 8; ++v) {
        aMsk[v] = 0.f; aMM[v] = 0.f; aMI[v] = 0.f;
        ps[v] = pseudo[row0 + rh + v];
    }

    const float* __restrict__ arow = q + (size_t)(row0 + m) * NDIM + kh;
    for (int nt = 0; nt < 4; ++nt) {
        const int col0 = colbase + nt * 64;
        const float* brow[4];
#pragma unroll
        for (int t = 0; t < 4; ++t)
            brow[t] = Qc + (size_t)(col0 + t * 16 + m) * NDIM + kh;

        v8f acc[4] = {};
        v2f a_c = *(const v2f*)(arow);
        v2f b_c[4];
#pragma unroll
        for (int t = 0; t < 4; ++t) b_c[t] = *(const v2f*)(brow[t]);

#pragma unroll
        for (int k = 0; k < NDIM - 4; k += 4) {
            const v2f a_n = *(const v2f*)(arow + k + 4);
            v2f b_n[4];
#pragma unroll
            for (int t = 0; t < 4; ++t) b_n[t] = *(const v2f*)(brow[t] + k + 4);
#pragma unroll
            for (int t = 0; t < 4; ++t) acc[t] = wmma_f32(a_c, b_c[t], acc[t]);
            a_c = a_n;
#pragma unroll
            for (int t = 0; t < 4; ++t) b_c[t] = b_n[t];
        }
#pragma unroll
        for (int t = 0; t < 4; ++t) acc[t] = wmma_f32(a_c, b_c[t], acc[t]);

#pragma unroll
        for (int t = 0; t < 4; ++t) {
            const int col = col0 + t * 16 + m;
            const int lab = label[col];
#pragma unroll
            for (int v = 0; v < 8; ++v) {
                const float sim = acc[t][v];
                const float mae = sqrtf(fmaxf(2.f - 2.f * sim + 1e-6f, 0.f));
                float msk = (ps[v] == lab) ? 1.f : 0.f;
                float mm  = mae * msk;
                float mi  = mae - mm;
                // sum over the 16 columns held by this lane half
                for (int off = 1; off < 16; off <<= 1) {
                    msk += __shfl_xor(msk, off, 32);
                    mm  += __shfl_xor(mm,  off, 32);
                    mi  += __shfl_xor(mi,  off, 32);
                }
                aMsk[v] += msk; aMM[v] += mm; aMI[v] += mi;
            }
        }
    }
    if (m == 0) {   // lanes 0 and 16 hold rows rh..rh+7
#pragma unroll
        for (int v = 0; v < 8; ++v) {
            const int r = row0 + rh + v;
            atomicAdd(&rowacc[r * 3 + 0], aMsk[v]);
            atomicAdd(&rowacc[r * 3 + 1], aMM[v]);
            atomicAdd(&rowacc[r * 3 + 2], aMI[v]);
        }
    }
}

// ------------------------------ small kernels ------------------------------

__global__ void zero_f32(float* __restrict__ p, int n) {
    int i = blockIdx.x * blockDim.x + threadIdx.x;
    if (i < n) p[i] = 0.f;
}

// row L2 normalize, 128 cols, block = 128 threads, grid = rows
__global__ void l2norm_rows(float* __restrict__ X) {
    const int row = blockIdx.x, t = threadIdx.x;
    float v = X[(size_t)row * NDIM + t];
    __shared__ float s[128];
    s[t] = v * v; __syncthreads();
    for (int off = 64; off; off >>= 1) { if (t < off) s[t] += s[t + off]; __syncthreads(); }
    const float nrm = fmaxf(sqrtf(s[0]), 1e-12f);
    X[(size_t)row * NDIM + t] = v / nrm;
}

// centroid segment-sum: grid = NQ, block = 128
__global__ void segsum(const float* __restrict__ Qc, const int* __restrict__ label,
                       float* __restrict__ cent, float* __restrict__ counts) {
    const int k = blockIdx.x, d = threadIdx.x;
    const int lab = label[k];
    atomicAdd(&cent[(size_t)lab * NDIM + d], Qc[(size_t)k * NDIM + d]);
    if (d == 0) atomicAdd(&counts[lab], 1.f);
}

// cent = l2norm(cent / max(counts,1)); grid = NINFO, block = 128
__global__ void cent_norm(float* __restrict__ cent, const float* __restrict__ counts) {
    const int row = blockIdx.x, t = threadIdx.x;
    const float c = fmaxf(counts[row], 1.f);
    float v = cent[(size_t)row * NDIM + t] / c;
    __shared__ float s[128];
    s[t] = v * v; __syncthreads();
    for (int off = 64; off; off >>= 1) { if (t < off) s[t] += s[t + off]; __syncthreads(); }
    const float nrm = fmaxf(sqrtf(s[0]), 1e-12f);
    cent[(size_t)row * NDIM + t] = v / nrm;
}

// row argmax (ties -> lowest index); grid = rows, block = 256
__global__ void argmax_rows(const float* __restrict__ S, int* __restrict__ out, int N) {
    const int row = blockIdx.x, t = threadIdx.x;
    float best = -3.0e38f; int bi = 0;
    for (int j = t; j < N; j += 256) {
        const float v = S[(size_t)row * N + j];
        if (v > best) { best = v; bi = j; }
    }
    __shared__ float sv[256]; __shared__ int si[256];
    sv[t] = best; si[t] = bi; __syncthreads();
    for (int off = 128; off; off >>= 1) {
        if (t < off) {
            if (sv[t + off] > sv[t] || (sv[t + off] == sv[t] && si[t + off] < si[t])) {
                sv[t] = sv[t + off]; si[t] = si[t + off];
            }
        }
        __syncthreads();
    }
    if (t == 0) out[row] = si[0];
}

// l_pos into logits col 0, and labels = 0; grid = NB, block = 128
__global__ void lpos_labels(const float* __restrict__ q, const float* __restrict__ k,
                            float* __restrict__ out) {
    const int row = blockIdx.x, t = threadIdx.x;
    __shared__ float s[128];
    s[t] = q[(size_t)row * NDIM + t] * k[(size_t)row * NDIM + t];
    __syncthreads();
    for (int off = 64; off; off >>= 1) { if (t < off) s[t] += s[t + off]; __syncthreads(); }
    if (t == 0) out[(long long)row * (NQ + 1)] = s[0] * INV_T;
    if (t == 1) out[(long long)NB * (NQ + 1) + row] = 0.f;   // labels
}

// mean cross-entropy over coarse logits; grid = NB, block = 128
__global__ void cls_loss_kernel(const float* __restrict__ cl,
                                const int* __restrict__ labs,
                                float* __restrict__ accum) {
    const int row = blockIdx.x, t = threadIdx.x;
    __shared__ float s[128];
    const float v = (t < NCOARSE) ? cl[(size_t)row * NCOARSE + t] : -3.0e38f;
    s[t] = v; __syncthreads();
    for (int off = 64; off; off >>= 1) { if (t < off) s[t] = fmaxf(s[t], s[t + off]); __syncthreads(); }
    const float mx = s[0]; __syncthreads();
    s[t] = (t < NCOARSE) ? expf(v - mx) : 0.f; __syncthreads();
    for (int off = 64; off; off >>= 1) { if (t < off) s[t] += s[t + off]; __syncthreads(); }
    if (t == 0) {
        const float lse = mx + logf(s[0]);
        const float lp = cl[(size_t)row * NCOARSE + labs[row]] - lse;
        atomicAdd(accum, -lp * (1.f / (float)NB));
    }
}

// dino finalize; 1 block x 1024 threads
__global__ void dino_final(const float* __restrict__ rowacc, float* __restrict__ out) {
    const int t = threadIdx.x;
    const float maskS = rowacc[t * 3 + 0];
    const float mmS   = rowacc[t * 3 + 1];
    const float miS   = rowacc[t * 3 + 2];
    const float eps = 1e-6f;
    __shared__ float s1[1024], s2[1024];
    s1[t] = mmS / (maskS + eps);
    s2[t] = miS / (((float)NQ - maskS) + eps);
    __syncthreads();
    for (int off = 512; off; off >>= 1) {
        if (t < off) { s1[t] += s1[t + off]; s2[t] += s2[t + off]; }
        __syncthreads();
    }
    if (t == 0) *out = s1[0] / (float)NB + (2.f - s2[0] / (float)NB);
}

// ---------------------------------------------------------------------------

extern "C" void kernel_launch(void* const* d_in, const int* in_sizes, int n_in,
                              void* d_out, int out_size, void* d_ws, size_t ws_size,
                              hipStream_t stream) {
    const float* h_q   = (const float*)d_in[0];
    const float* h_k   = (const float*)d_in[1];
    const float* W_cls = (const float*)d_in[2];
    const float* b_cls = (const float*)d_in[3];
    const float* W_p1  = (const float*)d_in[4];
    const float* b_p1  = (const float*)d_in[5];
    const float* W_p2  = (const float*)d_in[6];
    const float* b_p2  = (const float*)d_in[7];
    const float* Wk_p1 = (const float*)d_in[8];
    const float* bk_p1 = (const float*)d_in[9];
    const float* Wk_p2 = (const float*)d_in[10];
    const float* bk_p2 = (const float*)d_in[11];
    const float* queue_emb  = (const float*)d_in[12];
    const float* queue_copy = (const float*)d_in[13];
    const int*   info_label = (const int*)d_in[14];
    const int*   coarse_lab = (const int*)d_in[15];
    float* out = (float*)d_out;

    // workspace layout (floats)
    float* ws = (float*)d_ws;
    size_t off = 0;
    float* tmp_q  = ws + off; off += (size_t)NB * NEMBED;       // 2M
    float* tmp_k  = ws + off; off += (size_t)NB * NEMBED;       // 2M
    float* qv     = ws + off; off += (size_t)NB * NDIM;
    float* kv     = ws + off; off += (size_t)NB * NDIM;
    float* cent   = ws + off; off += (size_t)NINFO * NDIM;
    float* counts = ws + off; off += NINFO;
    float* scores = ws + off; off += (size_t)NB * NINFO;        // 1M
    float* coarse = ws + off; off += (size_t)NB * NCOARSE;
    float* rowacc = ws + off; off += (size_t)NB * 3;
    int*   pseudo = (int*)(ws + off); off += NB;

    const long long OFF_LABELS = (long long)NB * (NQ + 1);
    const long long OFF_DINO   = OFF_LABELS + NB;
    const long long OFF_CLS    = OFF_DINO + 1;

    // --- zero accumulators (every call: deterministic under graph replay) ---
    zero_f32<<<(NINFO * NDIM + 255) / 256, 256, 0, stream>>>(cent, NINFO * NDIM);
    zero_f32<<<(NINFO + 255) / 256, 256, 0, stream>>>(counts, NINFO);
    zero_f32<<<(NB * 3 + 255) / 256, 256, 0, stream>>>(rowacc, NB * 3);
    zero_f32<<<1, 1, 0, stream>>>(out + OFF_CLS, 1);

    // --- MLP layer 1 (ReLU), both branches:  [1024,2048]x[2048,2048] ---
    {
        dim3 grid(NEMBED / 512, NB / 32);          // MT=2: 32 rows per wave tile
        gemm_nn_t<NEMBED, NEMBED, true, 2, false, false>
            <<<grid, 256, 0, stream>>>(h_q, W_p1, b_p1, tmp_q);
        gemm_nn_t<NEMBED, NEMBED, true, 2, false, false>
            <<<grid, 256, 0, stream>>>(h_k, Wk_p1, bk_p1, tmp_k);
    }
    // --- MLP layer 2: [1024,2048]x[2048,128] (wave guard only, pipelined) ---
    {
        dim3 grid(1, NB / 16);
        gemm_nn_t<NEMBED, NDIM, false, 1, true, false>
            <<<grid, 256, 0, stream>>>(tmp_q, W_p2, b_p2, qv);
        gemm_nn_t<NEMBED, NDIM, false, 1, true, false>
            <<<grid, 256, 0, stream>>>(tmp_k, Wk_p2, bk_p2, kv);
    }
    // --- coarse classifier: [1024,2048]x[2048,100] (element guards) ---
    {
        dim3 grid(1, NB / 16);
        gemm_nn_t<NEMBED, NCOARSE, false, 1, true, true>
            <<<grid, 256, 0, stream>>>(h_q, W_cls, b_cls, coarse);
    }
    // --- L2 normalize q, k ---
    l2norm_rows<<<NB, 128, 0, stream>>>(qv);
    l2norm_rows<<<NB, 128, 0, stream>>>(kv);

    // --- dino centroids ---
    segsum<<<NQ, 128, 0, stream>>>(queue_copy, info_label, cent, counts);
    cent_norm<<<NINFO, 128, 0, stream>>>(cent, counts);

    // --- pseudo labels: scores = q @ cent^T, then row argmax ---
    {
        dim3 grid(NINFO / 512, NB / 32);           // MT=2
        gemm_nt_t<NDIM, 2><<<grid, 256, 0, stream>>>(qv, cent, scores,
                                                     1.0f, (long long)NINFO, 0);
    }
    argmax_rows<<<NB, 256, 0, stream>>>(scores, pseudo, NINFO);

    // --- logits: l_neg = (q @ queue^T)/T into d_out cols 1..K ---
    {
        dim3 grid(NQ / 512, NB / 32);              // MT=2
        gemm_nt_t<NDIM, 2><<<grid, 256, 0, stream>>>(qv, queue_emb, out, INV_T,
                                                     (long long)(NQ + 1), 1);
    }
    // --- l_pos into col 0, labels = 0 ---
    lpos_labels<<<NB, 128, 0, stream>>>(qv, kv, out);

    // --- dino fused sim/mae/mask accumulation + finalize ---
    {
        dim3 grid(NQ / 2048, NB / 16);             // 8 waves x 256 cols per block
        dino_accum<<<grid, 256, 0, stream>>>(qv, queue_copy, info_label, pseudo, rowacc);
    }
    dino_final<<<1, 1024, 0, stream>>>(rowacc, out + OFF_DINO);

    // --- classifier cross-entropy ---
    cls_loss_kernel<<<NB, 128, 0, stream>>>(coarse, coarse_lab, out + OFF_CLS);
}